// ModelNew_33028298506736
// MI455X (gfx1250) — compile-verified
//
#include <hip/hip_runtime.h>

#define N_TOK   1024
#define CZ      128
#define IN_DIMS 139      // 4*32 + 2*2 + 7
#define RMAX    32
#define SMAX    2
#define POS_OFF  0       // rows 0..65
#define TOK_OFF  66      // rows 66..131
#define ENT_ROW  132     // row 132
#define CH_OFF   133     // rows 133..138

#define ROWS_PER_BLOCK 2
#define NBLOCKS (N_TOK / ROWS_PER_BLOCK)   // 512
#define NTHREADS 256

typedef _Float16 half4 __attribute__((ext_vector_type(4)));
typedef float    f32x4 __attribute__((ext_vector_type(4)));
typedef int      i32x4 __attribute__((ext_vector_type(4)));
typedef int      i32x2 __attribute__((ext_vector_type(2)));

// Address-space-qualified vector pointers for the async builtin
typedef __attribute__((address_space(1))) i32x4 gas_i32x4;   // global
typedef __attribute__((address_space(3))) i32x4 las_i32x4;   // LDS

#if defined(__has_builtin)
#if __has_builtin(__builtin_amdgcn_global_load_async_to_lds_b128)
#define HAVE_ASYNC_LDS 1
#endif
#endif

__device__ __forceinline__ int clampi(int v, int lo, int hi) {
    v = v < lo ? lo : v;
    return v > hi ? hi : v;
}

__global__ __launch_bounds__(NTHREADS) void relpos_kernel(
    const int* __restrict__ asym_id,
    const int* __restrict__ residue_index,
    const int* __restrict__ entity_id,
    const int* __restrict__ token_index,
    const int* __restrict__ sym_id,
    const float* __restrict__ weight,
    float* __restrict__ out)
{
    // One fused f16 table, transposed to [k][c] (pos | token | entity | chain)
    __shared__ alignas(16) _Float16 tab[IN_DIMS * CZ];   // 35,584 B
    // SoA staging target for the async loads (prologue only)
    __shared__ alignas(16) int metaSoA[5 * N_TOK];       // 20,480 B
    // Packed per-token metadata: .x = asym | ent<<10 | sym<<20 ; .y = res | tok<<16
    __shared__ alignas(16) i32x2 metaPk[N_TOK];          //  8,192 B

    const int tid = threadIdx.x;

    // ---- stage metadata into LDS (async global->LDS, 16 B per lane) ----
    const int* srcs[5] = { asym_id, residue_index, entity_id, token_index, sym_id };
#ifdef HAVE_ASYNC_LDS
    #pragma unroll
    for (int a = 0; a < 5; ++a) {
        const int* g = srcs[a] + tid * 4;                 // 256 lanes * 16B = 4 KB
        int*       l = &metaSoA[a * N_TOK + tid * 4];
        __builtin_amdgcn_global_load_async_to_lds_b128(
            (gas_i32x4*)g, (las_i32x4*)l, /*imm offset*/0, /*cpol*/0);
    }
#else
    for (int a = 0; a < 5; ++a) {
        i32x4 v = *(const i32x4*)(srcs[a] + tid * 4);
        *(i32x4*)&metaSoA[a * N_TOK + tid * 4] = v;
    }
#endif

    // ---- stage weight table: f32 -> f16 (matches reference astype(f16)),
    //      transpose from [c][k] to [k][c]. Batched loads so they clause. ----
    {
        const int TOTAL = IN_DIMS * CZ;        // 17792 = 69.5 * 256
        #pragma unroll
        for (int grp = 0; grp < 5; ++grp) {
            float v[14];
            #pragma unroll
            for (int g = 0; g < 14; ++g) {
                int idx = tid + (grp * 14 + g) * NTHREADS;
                if (idx < TOTAL) {
                    int k = idx >> 7;          // feature row
                    int c = idx & 127;         // channel
                    v[g] = weight[c * IN_DIMS + k];
                }
            }
            #pragma unroll
            for (int g = 0; g < 14; ++g) {
                int idx = tid + (grp * 14 + g) * NTHREADS;
                if (idx < TOTAL) tab[idx] = (_Float16)v[g];
            }
        }
    }

#ifdef HAVE_ASYNC_LDS
    __builtin_amdgcn_s_wait_asynccnt(0);   // our async writes visible before barrier
#endif
    __syncthreads();

    // ---- pack 5 SoA ints per token into one int2 (4 tokens per thread) ----
    {
        const int j0 = tid * 4;
        i32x4 a = *(const i32x4*)&metaSoA[0 * N_TOK + j0];
        i32x4 r = *(const i32x4*)&metaSoA[1 * N_TOK + j0];
        i32x4 e = *(const i32x4*)&metaSoA[2 * N_TOK + j0];
        i32x4 t = *(const i32x4*)&metaSoA[3 * N_TOK + j0];
        i32x4 s = *(const i32x4*)&metaSoA[4 * N_TOK + j0];
        __syncthreads();                   // all reads done before repack
        #pragma unroll
        for (int q = 0; q < 4; ++q) {
            i32x2 p;
            p.x = a[q] | (e[q] << 10) | (s[q] << 20);
            p.y = r[q] | (t[q] << 16);
            metaPk[j0 + q] = p;
        }
    }
    __syncthreads();

    const int wave = tid >> 5;
    const int lane = tid & 31;
    const int c    = lane * 4;          // 4 channels per lane -> b128 store

    // ---- this wave's row i and its scalars ----
    const int i  = blockIdx.x * ROWS_PER_BLOCK + (wave >> 2);   // 4 waves per row
    const i32x2 pi = metaPk[i];
    const int ai = pi.x & 0x3ff;
    const int ei = (pi.x >> 10) & 0x3ff;
    const int si = (pi.x >> 20) & 0x3ff;
    const int ri = pi.y & 0xffff;
    const int ti = (pi.y >> 16) & 0xffff;

    const f32x4 fent = __builtin_convertvector(*(const half4*)&tab[ENT_ROW * CZ + c], f32x4);

    // 4 waves stride over j for this row
    for (int j = (wave & 3); j < N_TOK; j += 4) {
        const i32x2 pj = metaPk[j];                         // one ds_load_b64
        const int aj = pj.x & 0x3ff;
        const int ej = (pj.x >> 10) & 0x3ff;
        const int sj = (pj.x >> 20) & 0x3ff;
        const int rj = pj.y & 0xffff;
        const int tj = (pj.y >> 16) & 0xffff;

        const bool same_chain = (ai == aj);
        const bool same_res   = (ri == rj);
        const bool same_ent   = (ei == ej);

        int dres = clampi(ri - rj + RMAX, 0, 2 * RMAX);
        if (!same_chain) dres = 2 * RMAX + 1;

        int dtok = clampi(ti - tj + RMAX, 0, 2 * RMAX);
        if (!(same_chain && same_res)) dtok = 2 * RMAX + 1;

        int dch = clampi(si - sj + SMAX, 0, 2 * SMAX);
        if (!same_ent) dch = 2 * SMAX + 1;

        f32x4 acc = __builtin_convertvector(*(const half4*)&tab[(POS_OFF + dres) * CZ + c], f32x4);
        acc += __builtin_convertvector(*(const half4*)&tab[(TOK_OFF + dtok) * CZ + c], f32x4);
        acc += __builtin_convertvector(*(const half4*)&tab[(CH_OFF  + dch ) * CZ + c], f32x4);
        if (same_ent) acc += fent;

        // 512 MiB of output: stream past L2 with a nontemporal b128 store
        f32x4* dst = (f32x4*)&out[((size_t)i * N_TOK + (size_t)j) * CZ + c];
        __builtin_nontemporal_store(acc, dst);
    }
}

extern "C" void kernel_launch(void* const* d_in, const int* in_sizes, int n_in,
                              void* d_out, int out_size, void* d_ws, size_t ws_size,
                              hipStream_t stream) {
    (void)in_sizes; (void)n_in; (void)d_ws; (void)ws_size; (void)out_size;
    const int*   asym_id       = (const int*)d_in[0];
    const int*   residue_index = (const int*)d_in[1];
    const int*   entity_id     = (const int*)d_in[2];
    const int*   token_index   = (const int*)d_in[3];
    const int*   sym_id        = (const int*)d_in[4];
    const float* weight        = (const float*)d_in[5];
    float*       out           = (float*)d_out;

    relpos_kernel<<<dim3(NBLOCKS), dim3(NTHREADS), 0, stream>>>(
        asym_id, residue_index, entity_id, token_index, sym_id, weight, out);
}